// RoIAlignAvg_4174708211917
// MI455X (gfx1250) — compile-verified
//
#include <hip/hip_runtime.h>
#include <hip/hip_bf16.h>

// RoIAlign (8x8 samples) + 2x2/stride-1 avg pool for MI455X (gfx1250, wave32).
//
// L2-bound gather kernel: features (164MB) fit the 192MB L2. 1 workgroup
// (256 thr = 8 wave32) per roi; lanes <-> sample points (spatially adjacent
// addresses); channels unrolled x8 per thread (one addr pair + imm offsets).
// Bilinear samples computed once, staged in double-buffered LDS (1 barrier/iter).
// 2x2 pooling = constant GEMM Y(48x16) = A(48x64) * X(64x16) on the tensor core
// (v_wmma_f32_16x16x4_f32, 8 K-chunks per 16-row M-block); the single leftover
// output row (m=48) is done by the mb==3 wave pair with plain VALU.

typedef float v8f __attribute__((ext_vector_type(8)));
typedef float v2f __attribute__((ext_vector_type(2)));

namespace {
constexpr int kH = 200, kW = 200, kC = 256;
constexpr int kHW = kH * kW;          // 40000
constexpr int kOutPerRoi = kC * 49;   // 12544
constexpr int kStride = 68;           // LDS row stride (floats): 8B-aligned float2
                                      // loads, (4*lane + kk) bank pattern -> conflict-free
}

__device__ __forceinline__ float pool_weight(int m, int p) {
  // A[m][p]: 0.25 if sample point p=(ph,pw) feeds pooled output m=(i,j).
  if (m >= 49) return 0.0f;
  const int i = m / 7, j = m % 7;
  const int ph = p >> 3, pw = p & 7;
  const bool hit = ((ph == i) | (ph == i + 1)) & ((pw == j) | (pw == j + 1));
  return hit ? 0.25f : 0.0f;
}

__global__ __launch_bounds__(256) void roialign_avg_kernel(
    const float* __restrict__ feat, const float* __restrict__ rois,
    const float* __restrict__ scale_ptr, float* __restrict__ out) {
  const int n = blockIdx.x;
  const int t = threadIdx.x;

  // ---- per-point bilinear setup (exactly mirrors the reference math) ----
  const int p  = t & 63;        // sample point 0..63
  const int cg = t >> 6;        // 0..3: channel sub-group for the gather phase
  const int ph = p >> 3, pw = p & 7;

  const float scale = *scale_ptr;
  const float* roi = rois + (size_t)n * 5;
  const int   b  = (int)roi[0];
  const float x1 = roi[1] * scale, y1 = roi[2] * scale;
  const float x2 = roi[3] * scale, y2 = roi[4] * scale;
  const float bin_h = fmaxf(y2 - y1 + 1.0f, 0.0f) * (1.0f / 7.0f);
  const float bin_w = fmaxf(x2 - x1 + 1.0f, 0.0f) * (1.0f / 7.0f);
  const float h = y1 + (float)ph * bin_h;
  const float w = x1 + (float)pw * bin_w;
  const float hs = fminf(floorf(h), (float)(kH - 2));
  const float ws = fminf(floorf(w), (float)(kW - 2));
  const float hr = h - hs, wr = w - ws;   // reference keeps hr/wr from pre-clip floor
  const int hi = (int)fmaxf(hs, 0.0f);
  const int wi = (int)fmaxf(ws, 0.0f);
  const bool valid = (h >= 0.0f) & (h < (float)kH) & (w >= 0.0f) & (w < (float)kW);
  const float vm  = valid ? 1.0f : 0.0f;
  const float w00 = (1.0f - hr) * (1.0f - wr) * vm;
  const float w01 = (1.0f - hr) * wr * vm;
  const float w10 = hr * (1.0f - wr) * vm;
  const float w11 = hr * wr * vm;
  const float* fbase = feat + (size_t)b * (kC * kHW) + (size_t)hi * kW + wi;

  // ---- per-wave pooling setup ----
  const int wave = t >> 5;        // 0..7
  const int lane = t & 31;
  const int mb   = __builtin_amdgcn_readfirstlane(wave & 3);  // wave-uniform M-block
  const int chh  = wave >> 2;     // channel half of the 32-ch tile
  const int mrow = mb * 16 + (lane & 15);
  // 8-chunk K window per M-block (covers all points its output rows touch)
  const int kstart = __builtin_amdgcn_readfirstlane(mb == 0 ? 0 : (mb == 1 ? 16 : 32));
  const int khalf  = (lane >> 4) << 1;   // lanes16-31 hold K=2,3
  const int hi8    = (lane >> 4) << 3;   // D-row offset for upper lane half

  // Constant A fragments (same for every roi/channel tile), ISA 16x4 f32 layout:
  // VGPR0 = K0 | K2 across lane halves, VGPR1 = K1 | K3.
  v2f afrag[8];
#pragma unroll
  for (int s = 0; s < 8; ++s) {
    const int p0 = kstart + s * 4 + khalf;
    afrag[s].x = pool_weight(mrow, p0);
    afrag[s].y = pool_weight(mrow, p0 + 1);
  }

  __shared__ __align__(16) float samp[2][32][kStride];  // double-buffered 32ch x 64pt

  const int chl = chh * 16 + (lane & 15);               // channel-in-tile for pooling
  float* const obase0 = out + (size_t)n * kOutPerRoi;

  for (int g = 0; g < 8; ++g) {  // 8 tiles of 32 channels
    float (*buf)[kStride] = samp[g & 1];

    // ---- gather phase: 8 channels per thread, 4-corner bilinear ----
    const int c0 = g * 32 + cg * 8;
#pragma unroll
    for (int u = 0; u < 8; ++u) {
      const float* fp = fbase + (size_t)(c0 + u) * kHW;
      const float f00 = fp[0], f01 = fp[1], f10 = fp[kW], f11 = fp[kW + 1];
      buf[cg * 8 + u][p] = w00 * f00 + w01 * f01 + w10 * f10 + w11 * f11;
    }
    __syncthreads();  // single barrier: double buffer covers the write-after-read side

    float* const oc = obase0 + (size_t)(g * 32 + chl) * 49;
    if (mb != 3) {
      // ---- pooling as GEMM on the tensor core (outputs m = mb*16 .. mb*16+15) ----
      v8f acc = {0.f, 0.f, 0.f, 0.f, 0.f, 0.f, 0.f, 0.f};
#pragma unroll
      for (int s = 0; s < 8; ++s) {
        const int kk = kstart + s * 4 + khalf;          // even -> 8B aligned
        const v2f bfrag = *(const v2f*)&buf[chl][kk];   // B layout: K0|K2 , K1|K3
        acc = __builtin_amdgcn_wmma_f32_16x16x4_f32(
            /*neg_a=*/false, afrag[s], /*neg_b=*/false, bfrag,
            /*c_mod=*/(short)0, acc, /*reuse_a=*/false, /*reuse_b=*/false);
      }
      // D layout: component v, lane l -> row v + 8*(l>=16), col l&15. All m < 48.
#pragma unroll
      for (int v = 0; v < 8; ++v) oc[mb * 16 + v + hi8] = acc[v];
    } else {
      // ---- leftover output row m=48 (i=6,j=6): points 54,55,62,63 ----
      if (lane < 16) {
        const float* r = buf[chl];
        oc[48] = (((r[54] + r[55]) + r[62]) + r[63]) * 0.25f;
      }
    }
  }
}

extern "C" void kernel_launch(void* const* d_in, const int* in_sizes, int n_in,
                              void* d_out, int out_size, void* d_ws, size_t ws_size,
                              hipStream_t stream) {
  (void)n_in; (void)out_size; (void)d_ws; (void)ws_size;
  const float* feat  = (const float*)d_in[0];
  const float* rois  = (const float*)d_in[1];
  const float* scale = (const float*)d_in[2];
  float* out = (float*)d_out;
  const int n_rois = in_sizes[1] / 5;  // 1024
  roialign_avg_kernel<<<n_rois, 256, 0, stream>>>(feat, rois, scale, out);
}